// GraphProjection_90297392431237
// MI455X (gfx1250) — compile-verified
//
#include <hip/hip_runtime.h>

typedef __attribute__((ext_vector_type(2))) float v2f;
typedef __attribute__((ext_vector_type(4))) float v4f;
typedef __attribute__((ext_vector_type(8))) float v8f;

#define CH 128
#define SZ 224
#define HW (SZ * SZ)
#define BIG 3.402823466e38f

struct F2 { float x, y; };   // natural align 4: rows are only 4B-aligned

// ---------------- Kernel 1: per-block min/max of transformed x,y ----------------
__global__ void __launch_bounds__(256)
xform_minmax(const float* __restrict__ pts, const float* __restrict__ Rm,
             const float* __restrict__ Tv, float* __restrict__ part, int N) {
    const float r0 = Rm[0], r1 = Rm[1], r3 = Rm[3], r4 = Rm[4], r6 = Rm[6], r7 = Rm[7];
    const float t0 = Tv[0], t1 = Tv[1];
    float mnx = BIG, mxx = -BIG, mny = BIG, mxy = -BIG;
    for (int i = blockIdx.x * blockDim.x + threadIdx.x; i < N;
         i += gridDim.x * blockDim.x) {
        float px = pts[3 * i], py = pts[3 * i + 1], pz = pts[3 * i + 2];
        float x = fmaf(pz, r6, fmaf(py, r3, fmaf(px, r0, t0)));
        float y = fmaf(pz, r7, fmaf(py, r4, fmaf(px, r1, t1)));
        mnx = fminf(mnx, x); mxx = fmaxf(mxx, x);
        mny = fminf(mny, y); mxy = fmaxf(mxy, y);
    }
    // wave32 butterfly reduce (ds_bpermute / swizzle path)
    for (int off = 16; off > 0; off >>= 1) {
        mnx = fminf(mnx, __shfl_xor(mnx, off, 32));
        mxx = fmaxf(mxx, __shfl_xor(mxx, off, 32));
        mny = fminf(mny, __shfl_xor(mny, off, 32));
        mxy = fmaxf(mxy, __shfl_xor(mxy, off, 32));
    }
    __shared__ float sm[8][4];
    int wave = threadIdx.x >> 5, lane = threadIdx.x & 31;
    if (lane == 0) { sm[wave][0] = mnx; sm[wave][1] = mxx; sm[wave][2] = mny; sm[wave][3] = mxy; }
    __syncthreads();
    if (threadIdx.x == 0) {
        float a = sm[0][0], b = sm[0][1], c = sm[0][2], d = sm[0][3];
        for (int j = 1; j < 8; ++j) {
            a = fminf(a, sm[j][0]); b = fmaxf(b, sm[j][1]);
            c = fminf(c, sm[j][2]); d = fmaxf(d, sm[j][3]);
        }
        part[blockIdx.x * 4 + 0] = a; part[blockIdx.x * 4 + 1] = b;
        part[blockIdx.x * 4 + 2] = c; part[blockIdx.x * 4 + 3] = d;
    }
}

// ---------------- Kernel 2: finalize {min_x, min_y, scale_x, scale_y} ----------------
__global__ void __launch_bounds__(256)
finalize_params(const float* __restrict__ part, float* __restrict__ prm, int nPart) {
    float mnx = BIG, mxx = -BIG, mny = BIG, mxy = -BIG;
    for (int i = threadIdx.x; i < nPart; i += blockDim.x) {
        mnx = fminf(mnx, part[4 * i + 0]); mxx = fmaxf(mxx, part[4 * i + 1]);
        mny = fminf(mny, part[4 * i + 2]); mxy = fmaxf(mxy, part[4 * i + 3]);
    }
    for (int off = 16; off > 0; off >>= 1) {
        mnx = fminf(mnx, __shfl_xor(mnx, off, 32));
        mxx = fmaxf(mxx, __shfl_xor(mxx, off, 32));
        mny = fminf(mny, __shfl_xor(mny, off, 32));
        mxy = fmaxf(mxy, __shfl_xor(mxy, off, 32));
    }
    __shared__ float sm[8][4];
    int wave = threadIdx.x >> 5, lane = threadIdx.x & 31;
    if (lane == 0) { sm[wave][0] = mnx; sm[wave][1] = mxx; sm[wave][2] = mny; sm[wave][3] = mxy; }
    __syncthreads();
    if (threadIdx.x == 0) {
        float a = sm[0][0], b = sm[0][1], c = sm[0][2], d = sm[0][3];
        for (int j = 1; j < 8; ++j) {
            a = fminf(a, sm[j][0]); b = fmaxf(b, sm[j][1]);
            c = fminf(c, sm[j][2]); d = fmaxf(d, sm[j][3]);
        }
        prm[0] = a;                       // min_x
        prm[1] = c;                       // min_y
        prm[2] = 223.99f / (b - a);       // scale_x
        prm[3] = 223.99f / (d - c);       // scale_y
    }
}

// ---------------- Kernel 3: WMMA point transform + bilinear gather ----------------
__global__ void __launch_bounds__(256)
gather_bilerp(const float* __restrict__ img, const float* __restrict__ pts,
              const float* __restrict__ Rm, const float* __restrict__ Tv,
              const float* __restrict__ prm, float* __restrict__ out, int N) {
    const int tid  = threadIdx.x;
    const int lane = tid & 31;
    const int m    = lane & 15;
    const int n0   = blockIdx.x * 256 + (tid & ~31);  // wave's base point

    // A-matrix (16x4 f32): row j = (R[0][j], R[1][j], R[2][j], T[j]), rows 2..15 = 0.
    // 32-bit A layout: lanes 0-15 hold (K=0,K=1) of row=lane; lanes 16-31 hold (K=2,K=3).
    float a0 = 0.f, a1 = 0.f;
    if      (lane == 0)  { a0 = Rm[0]; a1 = Rm[3]; }
    else if (lane == 1)  { a0 = Rm[1]; a1 = Rm[4]; }
    else if (lane == 16) { a0 = Rm[6]; a1 = Tv[0]; }
    else if (lane == 17) { a0 = Rm[7]; a1 = Tv[1]; }
    v2f A = {a0, a1};

    // B-matrix (4x16): column n = (x_n, y_n, z_n, 1). Two groups of 16 points per wave.
    int last = N - 1;
    int q0 = n0 + m;      if (q0 > last) q0 = last;
    int q1 = n0 + 16 + m; if (q1 > last) q1 = last;
    float p0x = pts[3 * q0], p0y = pts[3 * q0 + 1], p0z = pts[3 * q0 + 2];
    float p1x = pts[3 * q1], p1y = pts[3 * q1 + 1], p1z = pts[3 * q1 + 2];
    bool hi = lane >= 16;
    v2f B0 = {hi ? p0z : p0x, hi ? 1.0f : p0y};
    v2f B1 = {hi ? p1z : p1x, hi ? 1.0f : p1y};

    v8f Z = {};
    // D[j][n] = p_n[j]:  vgpr0/lane(n) = x_n, vgpr1/lane(n) = y_n  (lanes 0-15)
    v8f D0 = __builtin_amdgcn_wmma_f32_16x16x4_f32(false, A, false, B0, (short)0, Z, false, false);
    v8f D1 = __builtin_amdgcn_wmma_f32_16x16x4_f32(false, A, false, B1, (short)0, Z, false, false);

    float xA = __shfl(D0[0], m, 32), yA = __shfl(D0[1], m, 32);   // points n0+m
    float xB = __shfl(D1[0], m, 32), yB = __shfl(D1[1], m, 32);   // points n0+16+m
    float xr = hi ? xB : xA;
    float yr = hi ? yB : yA;

    const int n = n0 + lane;
    if (n >= N) return;

    // normalize, clamp (guards floor/ceil against cross-pass ulp drift)
    float x = (xr - prm[0]) * prm[2];
    float y = (yr - prm[1]) * prm[3];
    x = fminf(fmaxf(x, 0.0f), 223.99f);
    y = fminf(fmaxf(y, 0.0f), 223.99f);

    int x1i = (int)floorf(x);
    int y1i = (int)floorf(y);
    int x2i = min((int)ceilf(x), SZ - 1);
    int y2i = min((int)ceilf(y), SZ - 1);
    float x1f = (float)x1i, x2f = (float)x2i, y1f = (float)y1i, y2f = (float)y2i;
    float w11 = (x2f - x) * (y2f - y);
    float w12 = (x2f - x) * (y - y1f);
    float w21 = (x - x1f) * (y2f - y);
    float w22 = (x - x1f) * (y - y1f);

    // pair gather: load (img[row+yb], img[row+yb+1]); fold edge cases into weights
    int yb = min(y1i, SZ - 2);
    bool s1 = (y1i == yb);   // Q11 sits at q.x (else q.y)
    bool s2 = (y2i == yb);   // Q12 sits at q.x (else q.y)
    float wa1 = (s1 ? w11 : 0.f) + (s2 ? w12 : 0.f);
    float wb1 = (s1 ? 0.f : w11) + (s2 ? 0.f : w12);
    float wa2 = (s1 ? w21 : 0.f) + (s2 ? w22 : 0.f);
    float wb2 = (s1 ? 0.f : w21) + (s2 ? 0.f : w22);

    const float* r1p = img + x1i * SZ + yb;
    const float* r2p = img + x2i * SZ + yb;
    float* op = out + (size_t)n * CH;

    for (int c = 0; c < CH; c += 4) {
        v4f v;
        #pragma unroll
        for (int k = 0; k < 4; ++k) {
            const F2 qa = *(const F2*)(r1p + (c + k) * HW);
            const F2 qb = *(const F2*)(r2p + (c + k) * HW);
            v[k] = fmaf(qa.x, wa1, fmaf(qa.y, wb1, fmaf(qb.x, wa2, qb.y * wb2)));
        }
        // NT store: don't let the 256MB output stream evict the L2-resident feature map
        __builtin_nontemporal_store(v, (v4f*)(op + c));
    }
}

extern "C" void kernel_launch(void* const* d_in, const int* in_sizes, int n_in,
                              void* d_out, int out_size, void* d_ws, size_t ws_size,
                              hipStream_t stream) {
    const float* img = (const float*)d_in[0];
    const float* pts = (const float*)d_in[1];
    const float* Rm  = (const float*)d_in[2];
    const float* Tv  = (const float*)d_in[3];
    float* out = (float*)d_out;
    float* ws  = (float*)d_ws;

    int N = in_sizes[1] / 3;

    float* prm  = ws;       // 4 floats: min_x, min_y, scale_x, scale_y
    float* part = ws + 8;   // 256 blocks * 4 floats of partial min/max

    const int rblocks = 256;
    xform_minmax<<<rblocks, 256, 0, stream>>>(pts, Rm, Tv, part, N);
    finalize_params<<<1, 256, 0, stream>>>(part, prm, rblocks);

    int gblocks = (N + 255) / 256;
    gather_bilerp<<<gblocks, 256, 0, stream>>>(img, pts, Rm, Tv, prm, out, N);
}